// InnerCos_13091060318226
// MI455X (gfx1250) — compile-verified
//
#include <hip/hip_runtime.h>
#include <hip/hip_bf16.h>
#include <math.h>

typedef float v2f __attribute__((ext_vector_type(2)));
typedef float v8f __attribute__((ext_vector_type(8)));

#define EPS_NORM  1e-8f
#define CLAMP_MIN 1e-12f
#define CLAMP_MAX 1e12f

// ---------------------------------------------------------------------------
// Kernel 1: zero the output scalar and precompute nc2[c] = sum_k centers[c][k]^2
// grid = C blocks, block = 256 threads.
// ---------------------------------------------------------------------------
__global__ void centers_norm_kernel(const float* __restrict__ centers,
                                    float* __restrict__ nc2,
                                    float* __restrict__ out,
                                    int D) {
    const int c = blockIdx.x;
    const int t = threadIdx.x;
    float v = 0.0f;
    for (int i = t; i < D; i += blockDim.x) {
        const float x = centers[(size_t)c * D + i];
        v += x * x;
    }
    // wave32 reduction
    for (int off = 16; off > 0; off >>= 1)
        v += __shfl_down(v, off, 32);
    __shared__ float s[8];
    if ((t & 31) == 0) s[t >> 5] = v;
    __syncthreads();
    if (t == 0) {
        float tot = 0.0f;
        const int nw = (blockDim.x + 31) >> 5;
        for (int i = 0; i < nw; ++i) tot += s[i];
        nc2[c] = tot;
        if (c == 0) out[0] = 0.0f;   // zero accumulator before main kernel
    }
}

// ---------------------------------------------------------------------------
// Diagonal extraction helper: pick element r (0..7) of the C/D fragment.
// ---------------------------------------------------------------------------
__device__ __forceinline__ float sel8(v8f c, int r) {
    float v = c[0];
    if (r == 1) v = c[1];
    if (r == 2) v = c[2];
    if (r == 3) v = c[3];
    if (r == 4) v = c[4];
    if (r == 5) v = c[5];
    if (r == 6) v = c[6];
    if (r == 7) v = c[7];
    return v;
}

// ---------------------------------------------------------------------------
// Kernel 2: each wave processes one 16-row tile via V_WMMA_F32_16X16X4_F32.
//   acc_dot[m][n] = <x_m, c_{label[n]}>   (we keep the diagonal)
//   acc_nx [m][n] = <x_m, x_n>            (diagonal = ||x_m||^2)
// ---------------------------------------------------------------------------
__global__ void inner_cos_kernel(const float* __restrict__ x,
                                 const int*   __restrict__ labels,
                                 const float* __restrict__ centers,
                                 const float* __restrict__ nc2,
                                 float* __restrict__ out,
                                 int N, int D, float invN) {
    const int lane = threadIdx.x & 31;
    const int wave = threadIdx.x >> 5;
    const int lo   = lane & 15;
    const int hi   = lane >> 4;

    const int tile     = blockIdx.x * (blockDim.x >> 5) + wave;
    const int row_base = tile << 4;

    float contrib = 0.0f;

    if (row_base + 16 <= N) {            // wave-uniform: EXEC stays all-ones
        const int  row  = row_base + lo; // this lane's A-row / B-column
        const int  cidx = labels[row];   // gathered center index for column lo

        const float* __restrict__ xp = x       + (size_t)row  * D;
        const float* __restrict__ cp = centers + (size_t)cidx * D;
        const int koff = hi << 1;        // lanes 16-31 hold K+2, K+3

        v8f acc_dot = {};
        v8f acc_nx  = {};

        #pragma unroll 8
        for (int k = 0; k < D; k += 4) {
            const v2f a = *reinterpret_cast<const v2f*>(xp + k + koff);
            const v2f b = *reinterpret_cast<const v2f*>(cp + k + koff);
            // D = A(16x4) * B(4x16) + C, fp32 in / fp32 accumulate
            acc_dot = __builtin_amdgcn_wmma_f32_16x16x4_f32(
                false, a, false, b, (short)0, acc_dot, false, false);
            acc_nx  = __builtin_amdgcn_wmma_f32_16x16x4_f32(
                false, a, false, a, (short)0, acc_nx,  false, false);
        }

        // Diagonal (M==N==m): m in [0,8) -> lane m, VGPR m;
        //                     m in [8,16) -> lane m+16, VGPR m-8.
        const bool active = hi ? (lo >= 8) : (lo < 8);
        const int  r      = lo & 7;

        const float dd   = sel8(acc_dot, r);          // <x_m, c_{l_m}>
        const float dn   = sel8(acc_nx,  r);          // ||x_m||^2
        const float ncs  = nc2[cidx];                 // ||c_{l_m}||^2 (same row: lo)
        const float den  = fmaxf(sqrtf(dn * ncs), EPS_NORM);
        const float cosd = 1.0f - dd / den;
        const float cl   = fminf(fmaxf(cosd, CLAMP_MIN), CLAMP_MAX);
        contrib = active ? cl : 0.0f;
    }

    // wave32 reduction
    for (int off = 16; off > 0; off >>= 1)
        contrib += __shfl_down(contrib, off, 32);

    __shared__ float ssum[8];
    if (lane == 0) ssum[wave] = contrib;
    __syncthreads();
    if (threadIdx.x == 0) {
        float tot = 0.0f;
        const int nw = blockDim.x >> 5;
        for (int i = 0; i < nw; ++i) tot += ssum[i];
        atomicAdd(out, tot * invN);
    }
}

// ---------------------------------------------------------------------------
// Launcher
// ---------------------------------------------------------------------------
extern "C" void kernel_launch(void* const* d_in, const int* in_sizes, int n_in,
                              void* d_out, int out_size, void* d_ws, size_t ws_size,
                              hipStream_t stream) {
    const float* ref_emb = (const float*)d_in[0];
    const int*   labels  = (const int*)  d_in[1];
    const float* centers = (const float*)d_in[2];

    const int N = in_sizes[1];              // 262144
    const int D = in_sizes[0] / N;          // 256
    const int C = in_sizes[2] / D;          // 64

    float* out = (float*)d_out;
    float* nc2 = (float*)d_ws;              // C floats of scratch

    // Pass 1: center norms + zero the output accumulator.
    centers_norm_kernel<<<C, 256, 0, stream>>>(centers, nc2, out, D);

    // Pass 2: one 16-row tile per wave, 8 waves per 256-thread block.
    const int tiles  = (N + 15) >> 4;
    const int blocks = (tiles + 7) >> 3;
    inner_cos_kernel<<<blocks, 256, 0, stream>>>(
        ref_emb, labels, centers, nc2, out, N, D, 1.0f / (float)N);
}